// FNetEncoder_42374147342818
// MI455X (gfx1250) — compile-verified
//
#include <hip/hip_runtime.h>
#include <hip/hip_bf16.h>

// ---------------------------------------------------------------------------
// FNet encoder block for MI455X (gfx1250), fp32 end-to-end.
//   fft_real = Re(FFT2(x))  = seq-FFT( x @ CosD  - i * x @ SinD ).real
//   proj_in  = LN(x + fft_real)
//   h        = relu(proj_in @ W1 + b1)
//   proj_out = h @ W2 + b2
//   out      = LN(proj_in + proj_out)
// GEMMs: V_WMMA_F32_16X16X4_F32; tiles staged into LDS by the Tensor Data
// Mover (tensor_load_to_lds + s_wait_tensorcnt), double-buffered so the DMA
// of tile s+1 overlaps the WMMA work on tile s.  (Probe-confirmed: the TDM
// builtins resolve on the gfx1250 device pass of this toolchain.)
// Seq FFT: Stockham radix-2 in LDS (128KB of the 320KB WGP LDS).
// ---------------------------------------------------------------------------

typedef __attribute__((ext_vector_type(2))) float v2f;
typedef __attribute__((ext_vector_type(4))) float v4f;
typedef __attribute__((ext_vector_type(8))) float v8f;
typedef __attribute__((ext_vector_type(4))) unsigned int v4u;
typedef __attribute__((ext_vector_type(4))) int v4i;
typedef __attribute__((ext_vector_type(8))) int v8i;

#define TWO_PI 6.28318530717958647692f

#ifndef __has_builtin
#define __has_builtin(x) 0
#endif
// Device pass: builtins present -> TDM staging. Host pass (no amdgcn
// builtins) parses the fallback branch; it never codegens kernel bodies.
#if __has_builtin(__builtin_amdgcn_tensor_load_to_lds) && \
    __has_builtin(__builtin_amdgcn_s_wait_tensorcnt)
#define USE_TDM 1
#else
#define USE_TDM 0
#endif

// ---------------------------------------------------------------------------
// DFT matrix generation: CosD[d,d'] = cos(2*pi*d*d'/512), SinD = sin(...)
// ---------------------------------------------------------------------------
__global__ __launch_bounds__(256) void init_dft_kernel(float* __restrict__ cosD,
                                                       float* __restrict__ sinD) {
  int i = blockIdx.x * 256 + threadIdx.x;          // 512*512 entries
  int d  = i >> 9;
  int dp = i & 511;
  float ang = TWO_PI * (float)((d * dp) & 511) / 512.0f;  // mod 512 for precision
  float s, c;
  __sincosf(ang, &s, &c);
  cosD[i] = c;
  sinD[i] = s;
}

#if USE_TDM
// ---------------------------------------------------------------------------
// TDM 2D tile load: global tile (tile_d1 rows x tile_d0 dwords, row stride
// stride0 dwords) -> LDS at lds_off, optional LDS padding (ISA 8.3-8.5).
// ---------------------------------------------------------------------------
__device__ __forceinline__ void tdm_load_2d(unsigned lds_off, const void* gptr,
                                            unsigned tile_d0, unsigned tile_d1,
                                            unsigned tens_d0, unsigned tens_d1,
                                            unsigned long long stride0,
                                            unsigned pad_interval,
                                            unsigned pad_amount, int pad_en) {
  unsigned long long ga = (unsigned long long)gptr;
  v4u g0;
  g0.x = 1u;                                         // count=1, is_restore=0
  g0.y = lds_off;                                    // lds_addr (bytes)
  g0.z = (unsigned)(ga & 0xFFFFFFFFull);             // global_addr[31:0]
  g0.w = (unsigned)((ga >> 32) & 0x01FFFFFFull) | (2u << 30);  // [56:32], type=2
  // group1 (256b): data_size=2 (4B), pad ctl, dims, strides
  unsigned d0 = (2u << 16);
  if (pad_en) d0 |= (1u << 20) | (pad_interval << 22) | (pad_amount << 25);
  unsigned long long q0 = (unsigned long long)d0 |
                          ((unsigned long long)(tens_d0 & 0xFFFFu) << 48);
  unsigned long long q1 = (unsigned long long)(tens_d0 >> 16) |
                          ((unsigned long long)tens_d1 << 16) |
                          ((unsigned long long)(tile_d0 & 0xFFFFu) << 48);
  unsigned long long q2 = (unsigned long long)(tile_d1 & 0xFFFFu) |
                          ((stride0 & 0xFFFFFFFFull) << 32);
  unsigned long long q3 = (stride0 >> 32) & 0xFFFFull;   // dim1_stride=0 (2D)
  v8i g1;
  g1[0] = (int)(unsigned)q0; g1[1] = (int)(unsigned)(q0 >> 32);
  g1[2] = (int)(unsigned)q1; g1[3] = (int)(unsigned)(q1 >> 32);
  g1[4] = (int)(unsigned)q2; g1[5] = (int)(unsigned)(q2 >> 32);
  g1[6] = (int)(unsigned)q3; g1[7] = (int)(unsigned)(q3 >> 32);
  v4i z4 = {0, 0, 0, 0};
#if __clang_major__ >= 23
  v8i z8 = {0, 0, 0, 0, 0, 0, 0, 0};
  __builtin_amdgcn_tensor_load_to_lds(g0, g1, z4, z4, z8, 0);
#else
  __builtin_amdgcn_tensor_load_to_lds(g0, g1, z4, z4, 0);
#endif
}
#endif  // USE_TDM

// ---------------------------------------------------------------------------
// fp32 WMMA GEMM: Cout[M,N] = A[M,K] @ B[K,N]  (+bias[N]) (+res[M,N]) (relu?)
// 128x128 macro tile, 256 threads = 8 wave32; each wave: 32(M) x 64(N)
// via 2x4 grid of 16x16 fp32 WMMA accumulators, K stepped by 16 (4x K=4).
// LDS: sA[m][k] stride 20 (TDM pad 4 dw / 16 dw), sB[k][n] stride 128.
// TDM path double-buffers: tile s+1 DMA overlaps tile s WMMA.
// ---------------------------------------------------------------------------
__global__ __launch_bounds__(256) void gemm_wmma_f32(
    const float* __restrict__ A, const float* __restrict__ B,
    float* __restrict__ Cout, const float* __restrict__ bias,
    const float* __restrict__ res, int M, int N, int K, int relu) {
  __shared__ float sA[2][128 * 20];   // 128 rows x 16 k, stride 20
  __shared__ float sB[2][16 * 128];   // 16 k rows x 128 n

  const int tid  = threadIdx.x;
  const int lane = tid & 31;
  const int wave = tid >> 5;
  const int wm   = wave & 3;       // 4 waves along M
  const int wn   = wave >> 2;      // 2 waves along N
  const int mBase = blockIdx.y * 128;
  const int nBase = blockIdx.x * 128;

  v8f acc[2][4] = {};              // 2 m-tiles x 4 n-tiles

  const int mrow  = wm * 32 + (lane & 15);     // fragment row in macro tile
  const int ncol  = wn * 64 + (lane & 15);     // fragment col in macro tile
  const int khalf = (lane >> 4) * 2;           // lanes 16..31 hold K+2,K+3

  const int ksteps = K >> 4;

#if USE_TDM
  const unsigned ldsA = (unsigned)(size_t)(void*)&sA[0][0];
  const unsigned ldsB = (unsigned)(size_t)(void*)&sB[0][0];
  // Prologue: DMA tile 0 into buffer 0 (one wave drives the TDM).
  if (tid < 32) {
    tdm_load_2d(ldsA, A + (size_t)mBase * K, 16u, 128u,
                (unsigned)K, (unsigned)M, (unsigned long long)K,
                3u /*16 dw*/, 3u /*4 dw -> stride 20*/, 1);
    tdm_load_2d(ldsB, B + nBase, 128u, 16u,
                (unsigned)N, (unsigned)K, (unsigned long long)N,
                0u, 0u, 0);
  }
#endif

  for (int s = 0; s < ksteps; ++s) {
    const int buf = s & 1;
#if USE_TDM
    if (tid < 32) {
      if (s + 1 < ksteps) {
        // Issue tile s+1 into the other buffer, then wait until <=2
        // outstanding: in-order completion guarantees tile s has landed.
        const int k1 = (s + 1) << 4;
        const unsigned o = (unsigned)((s + 1) & 1);
        tdm_load_2d(ldsA + o * (128u * 20u * 4u),
                    A + (size_t)mBase * K + k1, 16u, 128u,
                    (unsigned)K, (unsigned)M, (unsigned long long)K, 3u, 3u, 1);
        tdm_load_2d(ldsB + o * (16u * 128u * 4u),
                    B + (size_t)k1 * N + nBase, 128u, 16u,
                    (unsigned)N, (unsigned)K, (unsigned long long)N, 0u, 0u, 0);
        __builtin_amdgcn_s_wait_tensorcnt((short)2);
      } else {
        __builtin_amdgcn_s_wait_tensorcnt((short)0);
      }
    }
    __syncthreads();                 // release consumers for buffer `buf`
#else
    // Fallback: VGPR-staged tile loads (single buffer).
    const int k0 = s << 4;
    #pragma unroll
    for (int q = tid; q < 512; q += 256) {     // A: 128x16 via float4
      int row = q >> 2, quad = q & 3;
      const float* gp = A + (size_t)(mBase + row) * K + (k0 + quad * 4);
      if (k0 + 16 < K) __builtin_prefetch(gp + 16, 0, 3);
      *(v4f*)(&sA[buf][0] + row * 20 + quad * 4) = *(const v4f*)gp;
    }
    #pragma unroll
    for (int q = tid; q < 512; q += 256) {     // B: 16x128 via float4
      int kk = q >> 5, nq = q & 31;
      const float* gp = B + (size_t)(k0 + kk) * N + (nBase + nq * 4);
      if (k0 + 16 < K) __builtin_prefetch(gp + (size_t)16 * N, 0, 3);
      *(v4f*)(&sB[buf][0] + kk * 128 + nq * 4) = *(const v4f*)gp;
    }
    __syncthreads();
#endif

    const float* cA = &sA[buf][0];
    const float* cB = &sB[buf][0];
    #pragma unroll
    for (int kk = 0; kk < 4; ++kk) {
      // A fragment (16x4 f32): lane l -> m = l&15, k = kk*4 + 2*(l>>4) + {0,1}
      v2f a0 = *(const v2f*)(cA + (mrow)      * 20 + kk * 4 + khalf);
      v2f a1 = *(const v2f*)(cA + (mrow + 16) * 20 + kk * 4 + khalf);
      #pragma unroll
      for (int j = 0; j < 4; ++j) {
        // B fragment (4x16 f32): lane l -> n = l&15, same k pair; the two
        // k's are 128 dwords apart -> ds_load_2addr_b32.
        const float* bp = cB + (kk * 4 + khalf) * 128 + ncol + j * 16;
        v2f b;
        b.x = bp[0];
        b.y = bp[128];
        acc[0][j] = __builtin_amdgcn_wmma_f32_16x16x4_f32(
            false, a0, false, b, (short)0, acc[0][j], false, false);
        acc[1][j] = __builtin_amdgcn_wmma_f32_16x16x4_f32(
            false, a1, false, b, (short)0, acc[1][j], false, false);
      }
    }
    __syncthreads();                 // all reads of `buf` done before reuse
  }

  // --- epilogue: C layout: lane l, elem e -> n = l&15, m = e + 8*(l>>4) ---
  const int nloc  = lane & 15;
  const int mhalf = (lane >> 4) * 8;
  #pragma unroll
  for (int i = 0; i < 2; ++i) {
    #pragma unroll
    for (int j = 0; j < 4; ++j) {
      int ng = nBase + wn * 64 + j * 16 + nloc;
      float bv = bias ? bias[ng] : 0.0f;
      #pragma unroll
      for (int e = 0; e < 8; ++e) {
        int mg = mBase + wm * 32 + i * 16 + mhalf + e;
        float v = acc[i][j][e] + bv;
        if (relu) v = fmaxf(v, 0.0f);
        size_t idx = (size_t)mg * N + ng;
        if (res) v += res[idx];
        Cout[idx] = v;
      }
    }
  }
}

// ---------------------------------------------------------------------------
// Seq-axis 8192-pt complex FFT per (batch, embed) column, Stockham radix-2.
// Input z = Cr - i*Si; writes Re(FFT(z)) in-place into Cr.
// LDS ping-pong: 2 x 8192 x float2 = 128KB (< 320KB WGP LDS).
// ---------------------------------------------------------------------------
__global__ __launch_bounds__(256) void fft_seq_kernel(float* __restrict__ Cr,
                                                      const float* __restrict__ Si,
                                                      int Sdim, int D) {
  __shared__ float2 buf0[8192];
  __shared__ float2 buf1[8192];
  const int tid = threadIdx.x;
  const int b   = blockIdx.x >> 9;          // D = 512
  const int dp  = blockIdx.x & 511;
  const size_t base = (size_t)b * Sdim * D + dp;

  for (int s = tid; s < 8192; s += 256) {
    size_t idx = base + (size_t)s * D;
    buf0[s] = make_float2(Cr[idx], -Si[idx]);   // z = C - i*Sn
  }
  __syncthreads();

  float2* src = buf0;
  float2* dst = buf1;
  for (int st = 0; st < 13; ++st) {
    const int Ls = 1 << st;
    for (int t = tid; t < 4096; t += 256) {
      int j = t & (Ls - 1);
      int k = t >> st;
      float sn, cs;
      __sincosf(-TWO_PI * (float)j / (float)(Ls << 1), &sn, &cs);
      float2 u = src[k * Ls + j];
      float2 v = src[k * Ls + j + 4096];
      float2 vw = make_float2(v.x * cs - v.y * sn, v.x * sn + v.y * cs);
      dst[2 * k * Ls + j]      = make_float2(u.x + vw.x, u.y + vw.y);
      dst[2 * k * Ls + j + Ls] = make_float2(u.x - vw.x, u.y - vw.y);
    }
    __syncthreads();
    float2* tmp = src; src = dst; dst = tmp;
  }

  for (int s = tid; s < 8192; s += 256) {
    Cr[base + (size_t)s * D] = src[s].x;        // real part only
  }
}

// ---------------------------------------------------------------------------
// out[row,:] = LN(a[row,:] (+ b[row,:]); gamma, beta), D = 512.
// One wave32 per row, 16 elems/lane, shfl_xor reduction.
// ---------------------------------------------------------------------------
__global__ __launch_bounds__(256) void add_ln_kernel(
    const float* __restrict__ a, const float* __restrict__ b,
    const float* __restrict__ gamma, const float* __restrict__ beta,
    float* __restrict__ out) {
  const int lane = threadIdx.x & 31;
  const int wave = threadIdx.x >> 5;
  const size_t row = (size_t)blockIdx.x * 8 + wave;
  const size_t roff = row * 512;

  float vals[16];
  float sum = 0.0f, sq = 0.0f;
  #pragma unroll
  for (int i = 0; i < 16; ++i) {
    int c = lane + i * 32;
    float v = a[roff + c];
    if (b) v += b[roff + c];
    vals[i] = v;
    sum += v;
    sq  += v * v;
  }
  #pragma unroll
  for (int off = 16; off >= 1; off >>= 1) {
    sum += __shfl_xor(sum, off, 32);
    sq  += __shfl_xor(sq,  off, 32);
  }
  const float mean = sum * (1.0f / 512.0f);
  const float var  = sq * (1.0f / 512.0f) - mean * mean;
  const float inv  = rsqrtf(var + 1e-3f);
  #pragma unroll
  for (int i = 0; i < 16; ++i) {
    int c = lane + i * 32;
    out[roff + c] = (vals[i] - mean) * inv * gamma[c] + beta[c];
  }
}

// ---------------------------------------------------------------------------
extern "C" void kernel_launch(void* const* d_in, const int* in_sizes, int n_in,
                              void* d_out, int out_size, void* d_ws, size_t ws_size,
                              hipStream_t stream) {
  (void)in_sizes; (void)n_in; (void)out_size; (void)ws_size;
  const float* x   = (const float*)d_in[0];
  const float* W1  = (const float*)d_in[1];
  const float* b1  = (const float*)d_in[2];
  const float* W2  = (const float*)d_in[3];
  const float* b2  = (const float*)d_in[4];
  const float* g1  = (const float*)d_in[5];
  const float* be1 = (const float*)d_in[6];
  const float* g2  = (const float*)d_in[7];
  const float* be2 = (const float*)d_in[8];
  float* out = (float*)d_out;

  const int Bn = 4, S = 8192, D = 512, F = 2048;
  const int M = Bn * S;                       // 32768 tokens
  const size_t ND = (size_t)M * D;            // 16M floats

  float* cosD = (float*)d_ws;                 // 512*512
  float* sinD = cosD + 512 * 512;             // 512*512
  float* Cbuf = sinD + 512 * 512;             // ND (cos-DFT, then Re(FFT2) in place)
  float* Snb  = Cbuf + ND;                    // ND (sin-DFT; reused for proj_out)
  float* proj = Snb + ND;                     // ND (proj_input)
  float* hbuf = proj + ND;                    // 8192 * F (FFN hidden, chunked)

  dim3 blk(256);

  // DFT matrices
  init_dft_kernel<<<1024, blk, 0, stream>>>(cosD, sinD);

  // D-axis DFT as two fp32 WMMA GEMMs: C = x @ CosD, Sn = x @ SinD
  gemm_wmma_f32<<<dim3(D / 128, M / 128), blk, 0, stream>>>(
      x, cosD, Cbuf, nullptr, nullptr, M, D, D, 0);
  gemm_wmma_f32<<<dim3(D / 128, M / 128), blk, 0, stream>>>(
      x, sinD, Snb, nullptr, nullptr, M, D, D, 0);

  // S-axis FFT per (batch, embed) column; real part written into Cbuf.
  fft_seq_kernel<<<Bn * D, blk, 0, stream>>>(Cbuf, Snb, S, D);

  // proj_input = LN(x + Re(FFT2))
  add_ln_kernel<<<M / 8, blk, 0, stream>>>(x, Cbuf, g1, be1, proj);

  // FFN, 8192-token chunks to bound hidden-activation scratch at 64MB.
  for (int c = 0; c < 4; ++c) {
    const float* pc = proj + (size_t)c * 8192 * D;
    float* oc = Snb + (size_t)c * 8192 * D;   // reuse Sn buffer for proj_out+res
    gemm_wmma_f32<<<dim3(F / 128, 8192 / 128), blk, 0, stream>>>(
        pc, W1, hbuf, b1, nullptr, 8192, F, D, 1);          // relu(x@W1+b1)
    gemm_wmma_f32<<<dim3(D / 128, 8192 / 128), blk, 0, stream>>>(
        hbuf, W2, oc, b2, pc, 8192, D, F, 0);               // h@W2+b2+proj_in
  }

  // out = LN(proj_input + proj_output)
  add_ln_kernel<<<M / 8, blk, 0, stream>>>(Snb, nullptr, g2, be2, out);
}